// TutelMixtureOfExperts_89223650607342
// MI455X (gfx1250) — compile-verified
//
#include <hip/hip_runtime.h>
#include <hip/hip_bf16.h>

// Problem constants (from reference)
#define Ntok 16384   // B*T
#define Dd   1024
#define Ee   8
#define Hh   2048
#define Cc   4096    // capacity = K*N/E
#define EPSg 1.1920929e-07f

typedef __attribute__((ext_vector_type(2))) float v2f;
typedef __attribute__((ext_vector_type(8))) float v8f;

// ---------------- gating: one wave per token ----------------
__global__ __launch_bounds__(256)
void moe_gate_kernel(const float* __restrict__ x,
                     const unsigned char* __restrict__ pmask,
                     const float* __restrict__ wg,
                     float* __restrict__ gate_out,   // [N, E]
                     int* __restrict__ idx1, int* __restrict__ idx2,
                     float* __restrict__ g1o, float* __restrict__ g2o) {
    int gtid = blockIdx.x * blockDim.x + threadIdx.x;
    int n = gtid >> 5;
    int lane = threadIdx.x & 31;
    if (n >= Ntok) return;

    const float* xr = x + (size_t)n * Dd;
    float acc[Ee];
#pragma unroll
    for (int e = 0; e < Ee; ++e) acc[e] = 0.f;
    for (int j = 0; j < Dd / 32; ++j) {
        float xv = xr[lane + 32 * j];
#pragma unroll
        for (int e = 0; e < Ee; ++e)
            acc[e] += xv * wg[(size_t)e * Dd + lane + 32 * j];
    }
#pragma unroll
    for (int e = 0; e < Ee; ++e) {
#pragma unroll
        for (int off = 16; off >= 1; off >>= 1)
            acc[e] += __shfl_xor(acc[e], off, 32);
    }
    // every lane now has all 8 logits; do softmax + top2 redundantly
    float mx = acc[0];
#pragma unroll
    for (int e = 1; e < Ee; ++e) mx = fmaxf(mx, acc[e]);
    float sc[Ee]; float s = 0.f;
#pragma unroll
    for (int e = 0; e < Ee; ++e) { sc[e] = expf(acc[e] - mx); s += sc[e]; }
    float inv = 1.f / s;
#pragma unroll
    for (int e = 0; e < Ee; ++e) sc[e] *= inv;
    int i1 = 0; float b1 = sc[0];
#pragma unroll
    for (int e = 1; e < Ee; ++e) if (sc[e] > b1) { b1 = sc[e]; i1 = e; }
    int i2 = -1; float b2 = -2.f;
#pragma unroll
    for (int e = 0; e < Ee; ++e) if (e != i1 && sc[e] > b2) { b2 = sc[e]; i2 = e; }
    float denom = fmaxf(b1 + b2, EPSg);
    if (lane == 0) {
        idx1[n] = i1; idx2[n] = i2;
        g1o[n] = b1 / denom; g2o[n] = b2 / denom;
    }
    if (lane < Ee)
        gate_out[(size_t)n * Ee + lane] = pmask[n] ? 0.f : sc[lane];
}

// ---------------- capacity scan: single block, order-preserving ----------------
#define NPT 64   // 256 threads * 64 tokens
__global__ __launch_bounds__(256)
void moe_scan_kernel(const int* __restrict__ idx1, const int* __restrict__ idx2,
                     int* __restrict__ slot2tok,
                     int* __restrict__ flat1, int* __restrict__ flat2) {
    __shared__ int cnt[256][Ee];
    __shared__ int tot1[Ee];
    int tid = threadIdx.x;
    for (int i = tid; i < Ee * Cc; i += 256) slot2tok[i] = -1;

    int base = tid * NPT;
    int c[Ee];
#pragma unroll
    for (int e = 0; e < Ee; ++e) c[e] = 0;
    for (int k = 0; k < NPT; ++k) c[idx1[base + k]]++;
#pragma unroll
    for (int e = 0; e < Ee; ++e) cnt[tid][e] = c[e];
    __syncthreads();
    if (tid < Ee) {
        int run = 0;
        for (int t = 0; t < 256; ++t) { int v = cnt[t][tid]; cnt[t][tid] = run; run += v; }
        tot1[tid] = run;
    }
    __syncthreads();
#pragma unroll
    for (int e = 0; e < Ee; ++e) c[e] = cnt[tid][e];
    for (int k = 0; k < NPT; ++k) {
        int n = base + k; int e = idx1[n]; int p = c[e]++;
        if (p < Cc) { slot2tok[e * Cc + p] = n; flat1[n] = e * Cc + p; }
        else flat1[n] = -1;
    }
    __syncthreads();
#pragma unroll
    for (int e = 0; e < Ee; ++e) c[e] = 0;
    for (int k = 0; k < NPT; ++k) c[idx2[base + k]]++;
#pragma unroll
    for (int e = 0; e < Ee; ++e) cnt[tid][e] = c[e];
    __syncthreads();
    if (tid < Ee) {
        int run = 0;
        for (int t = 0; t < 256; ++t) { int v = cnt[t][tid]; cnt[t][tid] = run; run += v; }
    }
    __syncthreads();
#pragma unroll
    for (int e = 0; e < Ee; ++e) c[e] = cnt[tid][e] + tot1[e];
    for (int k = 0; k < NPT; ++k) {
        int n = base + k; int e = idx2[n]; int p = c[e]++;
        if (p < Cc) { slot2tok[e * Cc + p] = n; flat2[n] = e * Cc + p; }
        else flat2[n] = -1;
    }
}

// ---------------- WMMA fp32 GEMM tiles ----------------
#define TM 128
#define TN 128
#define TK 32
#define LSTR 36   // padded LDS row stride (floats): conflict-free, 16B aligned

// GEMM1: H = relu(disp @ fc1_w^T + b1); A rows gathered via slot2tok
__global__ __launch_bounds__(256)
void moe_ffn1_kernel(const float* __restrict__ x, const int* __restrict__ slot2tok,
                     const float* __restrict__ fc1_w, const float* __restrict__ fc1_b,
                     float* __restrict__ Hbuf) {
    int e = blockIdx.z;
    int m0 = blockIdx.x * TM;   // over C
    int n0 = blockIdx.y * TN;   // over H
    __shared__ float As[TM * LSTR];
    __shared__ float Bs[TN * LSTR];
    int tid = threadIdx.x;
    int lane = tid & 31;
    int w = tid >> 5;
    int wm = w >> 2;     // 0..1
    int wn = w & 3;      // 0..3
    int half = lane >> 4;
    int l16 = lane & 15;

    const float* wB = fc1_w + (size_t)e * Hh * Dd;
    v8f acc[4][2];
    v8f vzero = {0.f,0.f,0.f,0.f,0.f,0.f,0.f,0.f};
#pragma unroll
    for (int i = 0; i < 4; ++i)
#pragma unroll
        for (int j = 0; j < 2; ++j) acc[i][j] = vzero;

    for (int k0 = 0; k0 < Dd; k0 += TK) {
        __syncthreads();
#pragma unroll
        for (int p = 0; p < 4; ++p) {           // A tile (gathered rows)
            int f4 = tid + p * 256;
            int row = f4 >> 3, c4 = f4 & 7;
            int t = slot2tok[e * Cc + m0 + row];
            float4 v = make_float4(0.f, 0.f, 0.f, 0.f);
            if (t >= 0) v = *(const float4*)(x + (size_t)t * Dd + k0 + c4 * 4);
            *(float4*)&As[row * LSTR + c4 * 4] = v;
        }
#pragma unroll
        for (int p = 0; p < 4; ++p) {           // B tile: fc1_w[n][k] row-major
            int f4 = tid + p * 256;
            int row = f4 >> 3, c4 = f4 & 7;
            float4 v = *(const float4*)(wB + (size_t)(n0 + row) * Dd + k0 + c4 * 4);
            *(float4*)&Bs[row * LSTR + c4 * 4] = v;
        }
        __syncthreads();
#pragma unroll
        for (int kk = 0; kk < TK; kk += 4) {
            int kc = kk + 2 * half;
            v2f a[4], b[2];
#pragma unroll
            for (int i = 0; i < 4; ++i)
                a[i] = *(const v2f*)&As[(wm * 64 + i * 16 + l16) * LSTR + kc];
#pragma unroll
            for (int j = 0; j < 2; ++j)
                b[j] = *(const v2f*)&Bs[(wn * 32 + j * 16 + l16) * LSTR + kc];
#pragma unroll
            for (int i = 0; i < 4; ++i)
#pragma unroll
                for (int j = 0; j < 2; ++j)
                    acc[i][j] = __builtin_amdgcn_wmma_f32_16x16x4_f32(
                        false, a[i], false, b[j], (short)0, acc[i][j], false, false);
        }
    }
#pragma unroll
    for (int i = 0; i < 4; ++i)
#pragma unroll
        for (int j = 0; j < 2; ++j) {
            int n = n0 + wn * 32 + j * 16 + l16;
            float bias = fc1_b[e * Hh + n];
#pragma unroll
            for (int r = 0; r < 8; ++r) {
                int m = m0 + wm * 64 + i * 16 + r + 8 * half;
                float vv = acc[i][j][r] + bias;
                Hbuf[((size_t)e * Cc + m) * Hh + n] = fmaxf(vv, 0.f);
            }
        }
}

// GEMM2: Y = H @ fc2_w + b2  (reduction over H; fc2_w is [H, D] K-major)
__global__ __launch_bounds__(256)
void moe_ffn2_kernel(const float* __restrict__ Hbuf,
                     const float* __restrict__ fc2_w, const float* __restrict__ fc2_b,
                     float* __restrict__ Ybuf) {
    int e = blockIdx.z;
    int m0 = blockIdx.x * TM;   // over C
    int n0 = blockIdx.y * TN;   // over D
    __shared__ float As[TM * LSTR];
    __shared__ float Bs[TN * LSTR];
    int tid = threadIdx.x;
    int lane = tid & 31;
    int w = tid >> 5;
    int wm = w >> 2;
    int wn = w & 3;
    int half = lane >> 4;
    int l16 = lane & 15;

    const float* wB = fc2_w + (size_t)e * Hh * Dd;
    v8f acc[4][2];
    v8f vzero = {0.f,0.f,0.f,0.f,0.f,0.f,0.f,0.f};
#pragma unroll
    for (int i = 0; i < 4; ++i)
#pragma unroll
        for (int j = 0; j < 2; ++j) acc[i][j] = vzero;

    for (int k0 = 0; k0 < Hh; k0 += TK) {
        __syncthreads();
#pragma unroll
        for (int p = 0; p < 4; ++p) {           // A tile from Hbuf (row-major [C,H])
            int f4 = tid + p * 256;
            int row = f4 >> 3, c4 = f4 & 7;
            float4 v = *(const float4*)(Hbuf + ((size_t)e * Cc + m0 + row) * Hh + k0 + c4 * 4);
            *(float4*)&As[row * LSTR + c4 * 4] = v;
        }
#pragma unroll
        for (int p = 0; p < 16; ++p) {          // B tile transpose: [k][n] -> Bs[n][k]
            int id = tid + p * 256;
            int n = id & 127;
            int k = id >> 7;
            Bs[n * LSTR + k] = wB[(size_t)(k0 + k) * Dd + n0 + n];
        }
        __syncthreads();
#pragma unroll
        for (int kk = 0; kk < TK; kk += 4) {
            int kc = kk + 2 * half;
            v2f a[4], b[2];
#pragma unroll
            for (int i = 0; i < 4; ++i)
                a[i] = *(const v2f*)&As[(wm * 64 + i * 16 + l16) * LSTR + kc];
#pragma unroll
            for (int j = 0; j < 2; ++j)
                b[j] = *(const v2f*)&Bs[(wn * 32 + j * 16 + l16) * LSTR + kc];
#pragma unroll
            for (int i = 0; i < 4; ++i)
#pragma unroll
                for (int j = 0; j < 2; ++j)
                    acc[i][j] = __builtin_amdgcn_wmma_f32_16x16x4_f32(
                        false, a[i], false, b[j], (short)0, acc[i][j], false, false);
        }
    }
#pragma unroll
    for (int i = 0; i < 4; ++i)
#pragma unroll
        for (int j = 0; j < 2; ++j) {
            int n = n0 + wn * 32 + j * 16 + l16;
            float bias = fc2_b[e * Dd + n];
#pragma unroll
            for (int r = 0; r < 8; ++r) {
                int m = m0 + wm * 64 + i * 16 + r + 8 * half;
                Ybuf[((size_t)e * Cc + m) * Dd + n] = acc[i][j][r] + bias;
            }
        }
}

// ---------------- combine: one block per token ----------------
__global__ __launch_bounds__(256)
void moe_combine_kernel(const float* __restrict__ Ybuf,
                        const int* __restrict__ flat1, const int* __restrict__ flat2,
                        const float* __restrict__ g1, const float* __restrict__ g2,
                        float* __restrict__ out) {
    int n = blockIdx.x;
    int d4 = threadIdx.x;       // 256 float4 = 1024 floats
    int f1 = flat1[n], f2 = flat2[n];
    float gg1 = g1[n], gg2 = g2[n];
    float4 y1 = make_float4(0.f, 0.f, 0.f, 0.f);
    float4 y2 = make_float4(0.f, 0.f, 0.f, 0.f);
    if (f1 >= 0) y1 = *(const float4*)(Ybuf + (size_t)f1 * Dd + d4 * 4);
    if (f2 >= 0) y2 = *(const float4*)(Ybuf + (size_t)f2 * Dd + d4 * 4);
    float4 o;
    o.x = gg1 * y1.x + gg2 * y2.x;
    o.y = gg1 * y1.y + gg2 * y2.y;
    o.z = gg1 * y1.z + gg2 * y2.z;
    o.w = gg1 * y1.w + gg2 * y2.w;
    *(float4*)(out + (size_t)n * Dd + d4 * 4) = o;
}

extern "C" void kernel_launch(void* const* d_in, const int* in_sizes, int n_in,
                              void* d_out, int out_size, void* d_ws, size_t ws_size,
                              hipStream_t stream) {
    const float* x      = (const float*)d_in[0];
    const unsigned char* pmask = (const unsigned char*)d_in[1];
    const float* wg     = (const float*)d_in[2];
    const float* fc1_w  = (const float*)d_in[3];
    const float* fc1_b  = (const float*)d_in[4];
    const float* fc2_w  = (const float*)d_in[5];
    const float* fc2_b  = (const float*)d_in[6];

    float* out      = (float*)d_out;                       // [N, D]
    float* gate_out = out + (size_t)Ntok * Dd;             // [N, E]

    char* ws = (char*)d_ws;
    int*   slot2tok = (int*)(ws + 0);                      // E*C ints
    int*   flat1    = (int*)(ws + 131072);
    int*   flat2    = (int*)(ws + 196608);
    float* g1       = (float*)(ws + 262144);
    float* g2       = (float*)(ws + 327680);
    int*   idx1     = (int*)(ws + 393216);
    int*   idx2     = (int*)(ws + 458752);
    float* Hbuf     = (float*)(ws + 524288);               // E*C*H floats = 256 MB
    float* Ybuf     = (float*)(ws + 524288 + (size_t)268435456); // E*C*D floats = 128 MB

    // 1. gating: one wave per token -> 2048 blocks of 256
    moe_gate_kernel<<<dim3((Ntok * 32) / 256), dim3(256), 0, stream>>>(
        x, pmask, wg, gate_out, idx1, idx2, g1, g2);

    // 2. capacity scan (single block, deterministic order)
    moe_scan_kernel<<<dim3(1), dim3(256), 0, stream>>>(idx1, idx2, slot2tok, flat1, flat2);

    // 3. expert FFN layer 1 (fused dispatch gather + bias + relu)
    moe_ffn1_kernel<<<dim3(Cc / TM, Hh / TN, Ee), dim3(256), 0, stream>>>(
        x, slot2tok, fc1_w, fc1_b, Hbuf);

    // 4. expert FFN layer 2 (+bias)
    moe_ffn2_kernel<<<dim3(Cc / TM, Dd / TN, Ee), dim3(256), 0, stream>>>(
        Hbuf, fc2_w, fc2_b, Ybuf);

    // 5. combine (writes every output element; no zero-init needed)
    moe_combine_kernel<<<dim3(Ntok), dim3(256), 0, stream>>>(
        Ybuf, flat1, flat2, g1, g2, out);
}